// CgpHmmLayer_51221779972207
// MI455X (gfx1250) — compile-verified
//
#include <hip/hip_runtime.h>
#include <math.h>
#include <stdint.h>

// CDNA5 async Global->LDS path (ASYNCcnt). Guarded: falls back to synchronous
// staging + global_prefetch_b8 if this toolchain doesn't declare the builtins.
#if __has_builtin(__builtin_amdgcn_global_load_async_to_lds_b32) && \
    __has_builtin(__builtin_amdgcn_s_wait_asynccnt)
#define HMM_ASYNC_OBS 1
#endif

typedef __attribute__((ext_vector_type(2))) float v2f;
typedef __attribute__((ext_vector_type(8))) float v8f;

typedef __attribute__((address_space(1))) int gas_int;  // global
typedef __attribute__((address_space(3))) int las_int;  // LDS

// Problem constants (from reference)
constexpr int kB = 128, kT = 8192, kS = 128, kE = 6;
// Tiling
constexpr int kBB    = 16;   // batch rows per block (one WMMA M-tile)
constexpr int kNW    = 8;    // waves per block (each owns a 16-state N-tile)
constexpr int kCHUNK = 128;  // obs staging chunk (time steps)
constexpr int kASTR  = 130;  // padded alpha row stride (floats)
constexpr int kOSTR  = 20;   // padded transposed-obs row stride (ints, keeps 16B align)

__device__ __forceinline__ void async_obs_b32(const int* g, int* l) {
#ifdef HMM_ASYNC_OBS
  // AS1/AS3 casts via uintptr (low 32 bits of a flat LDS address are the LDS offset)
  __builtin_amdgcn_global_load_async_to_lds_b32(
      (gas_int*)(uintptr_t)g,
      (las_int*)(uint32_t)(uintptr_t)l,
      0, 0);
#else
  (void)g; (void)l;
#endif
}

__global__ __launch_bounds__(256) void hmm_forward_wmma(
    const int*   __restrict__ obs,   // [kB][kT]
    const float* __restrict__ Iv,    // [kS]
    const float* __restrict__ A,     // [kS][kS]
    const float* __restrict__ Bm,    // [kS][kE]
    float*       __restrict__ out)   // [kB]
{
  __shared__ __align__(16) float alpha[2][kBB][kASTR]; // ping-pong unnormalized alpha
  __shared__ float Zpart[2][kBB][kNW];                 // per-wave partial normalizers
  __shared__ float BmL[kS][9];                         // emission table, padded
  __shared__ __align__(16) int obsT[2][kCHUNK][kOSTR]; // transposed, double-buffered obs

  const int tid  = threadIdx.x;
  const int lane = tid & 31;
  const int wid  = tid >> 5;
  const int b0   = blockIdx.x * kBB;
  const int n0   = wid * 16;
  const int lm   = lane & 15;
  const int hi   = lane >> 4;

  // ---- stage Bm -> LDS ----
  for (int idx = tid; idx < kS * kE; idx += 256) {
    int s = idx / kE, e = idx - s * kE;
    BmL[s][e] = Bm[idx];
  }
  // ---- stage obs chunk 0 (sync, transposed) ----
  {
    const int m = tid >> 4, c0 = (tid & 15) * 8;
    const int* src = &obs[(b0 + m) * kT + c0];
#pragma unroll
    for (int j = 0; j < 8; ++j) obsT[0][c0 + j][m] = src[j];
  }
#ifdef HMM_ASYNC_OBS
  // issue chunk 1 on the async engine; consumed (after wait) at t = kCHUNK
  {
    const int m = tid >> 4, c0 = (tid & 15) * 8;
    const int* src = &obs[(b0 + m) * kT + kCHUNK + c0];
#pragma unroll
    for (int j = 0; j < 8; ++j) async_obs_b32(src + j, &obsT[1][c0 + j][m]);
  }
#endif
  if (tid < kBB * kNW) Zpart[0][tid >> 3][tid & 7] = 0.0f;

  // ---- preload time-invariant transition-matrix fragments (WMMA B, 4x16 f32) ----
  v2f fb[32];
  {
    const int koff = hi * 2;
    const int n = n0 + lm;
#pragma unroll
    for (int kk = 0; kk < 32; ++kk) {
      const int k0 = kk * 4 + koff;
      v2f f;
      f.x = A[(k0    ) * kS + n];
      f.y = A[(k0 + 1) * kS + n];
      fb[kk] = f;
    }
  }
  __syncthreads();

  // ---- t = 0: alpha0 = I * em0 ----
  {
    const int m  = tid >> 4;
    const int s0 = (tid & 15) * 8;
    const int o  = obsT[0][0][m];
    float part = 0.0f;
#pragma unroll
    for (int j = 0; j < 8; ++j) {
      float v = Iv[s0 + j] * BmL[s0 + j][o];
      alpha[0][m][s0 + j] = v;
      part += v;
    }
    atomicAdd(&Zpart[0][m][tid & 7], part);
  }
  __syncthreads();

  float ll = 0.0f;  // wave 0, lanes 0..15 (batch row = lane)

  for (int t = 1; t < kT; ++t) {
    const int cb = (t - 1) & 1, nb = t & 1;
    const int tc = t & (kCHUNK - 1);
    const int buf = (t >> 7) & 1;

    if (tc == 0) {
#ifdef HMM_ASYNC_OBS
      __builtin_amdgcn_s_wait_asynccnt(0);  // chunk for this window has landed
      __syncthreads();
      if (t + kCHUNK < kT) {                // issue next chunk into the retired buffer
        const int m = tid >> 4, c0 = (tid & 15) * 8;
        const int* src = &obs[(b0 + m) * kT + t + kCHUNK + c0];
#pragma unroll
        for (int j = 0; j < 8; ++j) async_obs_b32(src + j, &obsT[buf ^ 1][c0 + j][m]);
      }
#else
      const int m = tid >> 4, c0 = (tid & 15) * 8;
      const int* src = &obs[(b0 + m) * kT + t + c0];
#pragma unroll
      for (int j = 0; j < 8; ++j) obsT[buf][c0 + j][m] = src[j];
      if (t + kCHUNK < kT)
        __builtin_prefetch(&obs[(b0 + m) * kT + t + kCHUNK + c0], 0, 1);
      __syncthreads();
#endif
    }

    const int koff = hi * 2;

    // Stage ALL alpha fragments first: 16x ds_load_2addr_b64 issue back-to-back,
    // hiding LDS latency behind the Z-sum + rcp below instead of per-pair waits.
    v2f afr[32];
#pragma unroll
    for (int kk = 0; kk < 32; ++kk)
      afr[kk] = *(const v2f*)&alpha[cb][lm][kk * 4 + koff];

    // Previous step's normalizer for row m = lm; wave 0 folds it into loglik.
    float Z = 0.0f;
#pragma unroll
    for (int w = 0; w < kNW; ++w) Z += Zpart[cb][lm][w];
    const float invZ = __builtin_amdgcn_rcpf(Z);
    if (wid == 0 && lane < 16) ll += __logf(Z);

    // tnew_tile = (alpha/Z) @ A : two interleaved accumulator chains
    v8f acc0 = {}, acc1 = {};
#pragma unroll
    for (int kk = 0; kk < 32; kk += 2) {
      v2f af0 = afr[kk];
      v2f af1 = afr[kk + 1];
      af0.x *= invZ; af0.y *= invZ;
      af1.x *= invZ; af1.y *= invZ;
      acc0 = __builtin_amdgcn_wmma_f32_16x16x4_f32(
          false, af0, false, fb[kk],     (short)0, acc0, false, false);
      acc1 = __builtin_amdgcn_wmma_f32_16x16x4_f32(
          false, af1, false, fb[kk + 1], (short)0, acc1, false, false);
    }

    // Observations for this half's 8 batch rows: contiguous -> 2x ds_load_b128
    const int4 q0 = *(const int4*)&obsT[buf][tc][hi * 8];
    const int4 q1 = *(const int4*)&obsT[buf][tc][hi * 8 + 4];
    const int oarr[8] = {q0.x, q0.y, q0.z, q0.w, q1.x, q1.y, q1.z, q1.w};

    // Emission multiply, per-row sum (wave32 halves reduce independently), stage alpha
    const int s = n0 + lm;
    float pr[8];
#pragma unroll
    for (int r = 0; r < 8; ++r) {
      const int m  = r + hi * 8;
      const float em = BmL[s][oarr[r]];
      float v = (acc0[r] + acc1[r]) * em;
      alpha[nb][m][s] = v;
      v += __shfl_xor(v, 1);
      v += __shfl_xor(v, 2);
      v += __shfl_xor(v, 4);
      v += __shfl_xor(v, 8);
      pr[r] = v;
    }
    if (lm == 0) {  // single exec toggle for all 8 partial-normalizer stores
#pragma unroll
      for (int r = 0; r < 8; ++r) Zpart[nb][r + hi * 8][wid] = pr[r];
    }
    __syncthreads();
  }

  // Fold in the final step's normalizer and write out.
  if (wid == 0 && lane < 16) {
    float Z = 0.0f;
#pragma unroll
    for (int w = 0; w < kNW; ++w) Z += Zpart[(kT - 1) & 1][lane][w];
    out[b0 + lane] = ll + __logf(Z);
  }
}

extern "C" void kernel_launch(void* const* d_in, const int* in_sizes, int n_in,
                              void* d_out, int out_size, void* d_ws, size_t ws_size,
                              hipStream_t stream) {
  const int*   obs = (const int*)d_in[0];
  const float* Iv  = (const float*)d_in[1];
  const float* A   = (const float*)d_in[2];
  const float* Bm  = (const float*)d_in[3];
  float*       out = (float*)d_out;
  hipLaunchKernelGGL(hmm_forward_wmma, dim3(kB / kBB), dim3(256), 0, stream,
                     obs, Iv, A, Bm, out);
}